// SimpleNN_27341761806597
// MI455X (gfx1250) — compile-verified
//
#include <hip/hip_runtime.h>

// MI455X / gfx1250, wave32. WMMA D = A x B + C with A 16x4, B 4x16, C/D 16x16 f32.
typedef float v2f __attribute__((ext_vector_type(2)));
typedef float v8f __attribute__((ext_vector_type(8)));

#define HID 64
#define TILES_PER_WAVE 32   // 512 rows per wave; amortizes weight setup

// VOP3 f32 ops with the free |src| abs modifier. Written as inline asm because
// the SLP vectorizer otherwise lowers fabsf to a separate v_and_b32 0x7fffffff
// feeding v_pk_fma_f32 (3 ops / 2 elements); this is 1 op per element.
__device__ __forceinline__ float mul_abs(float s, float z) {
  float d;
  asm("v_mul_f32 %0, %1, |%2|" : "=v"(d) : "v"(s), "v"(z));
  return d;
}
__device__ __forceinline__ float fma_abs(float s, float z, float acc) {
  float d;
  asm("v_fma_f32 %0, %1, |%2|, %3" : "=v"(d) : "v"(s), "v"(z), "v"(acc));
  return d;
}

template <int IDX>
__device__ __forceinline__ float xform(float v) {
  if (IDX == 0) return v;
  if (IDX == 1) return __logf(v);
  if (IDX == 2) return __expf(v);
  return __sinf(v);
}

// Straight-line tile loop; transform selected at compile time (gate idx is
// wave-uniform; dispatched once outside the loop).
template <int IDX>
__device__ __forceinline__ void run_tiles(
    const float* __restrict__ x, float* __restrict__ out,
    int tile0, int ntiles, int nrows,
    int lane, int n, int half,
    const v2f aop[4], const v8f cinit[4], const float sreg[4][8],
    float T0, float T1, float T2)
{
#pragma unroll 2            // overlap tile k+1 WMMA (XDL) with tile k abs-VALU
  for (int it = 0; it < TILES_PER_WAVE; ++it) {
    const int T = tile0 + it;
    if (T >= ntiles) break;                    // wave-uniform; EXEC stays full

    // ---- gather + transform 16 (a,p) pairs (halves duplicate -> cache hits)
    int r = T * 16 + n;                        // output row = b*5 + i
    const int rc = r < nrows ? r : (nrows - 1);
    const unsigned b  = (unsigned)rc / 5u;
    const unsigned i  = (unsigned)rc - 5u * b;
    const unsigned ad = b * 10u + i;
    const float xa = x[ad];
    const float xp = x[ad + 5u];
    __builtin_prefetch(x + ad + 40u, 0, 1);    // next tile's lines -> global_prefetch_b8

    const float a = xform<IDX>(xa);
    const float p = xform<IDX>(xp);

    // Stage-1 B operand: X^T, 4x16.  VGPR0: K0 row (lanes 0-15) / K2 pad (16-31);
    //                                VGPR1: K1 / K3 pad.
    v2f bop;
    bop[0] = half ? 0.0f : a;
    bop[1] = half ? 0.0f : p;

    // ---- stage 1: z'' = (w2.*W1)^T x X^T + w2.*b1   (4 x V_WMMA_F32_16X16X4_F32)
    v8f acc0 = __builtin_amdgcn_wmma_f32_16x16x4_f32(false, aop[0], false, bop, (short)0, cinit[0], false, false);
    v8f acc1 = __builtin_amdgcn_wmma_f32_16x16x4_f32(false, aop[1], false, bop, (short)0, cinit[1], false, false);
    v8f acc2 = __builtin_amdgcn_wmma_f32_16x16x4_f32(false, aop[2], false, bop, (short)0, cinit[2], false, false);
    v8f acc3 = __builtin_amdgcn_wmma_f32_16x16x4_f32(false, aop[3], false, bop, (short)0, cinit[3], false, false);

    // ---- stage 2:  sum_j w2_j*relu(z_j) = 0.5*sum_j z''_j + 0.5*sum_j s_j*|z''_j|
    // Linear half is precomputed (T0,T1,T2); abs half is exactly 1 VALU op per
    // element: v_fma_f32 dst, s, |z|, acc.
    float part[8];
#pragma unroll
    for (int vv = 0; vv < 8; ++vv) {
      float s = mul_abs(sreg[0][vv], acc0[vv]);
      s = fma_abs(sreg[1][vv], acc1[vv], s);
      s = fma_abs(sreg[2][vv], acc2[vv], s);
      s = fma_abs(sreg[3][vv], acc3[vv], s);
      part[vv] = s;
    }
    float q0 = part[0] + part[1];
    float q1 = part[2] + part[3];
    float q2 = part[4] + part[5];
    float q3 = part[6] + part[7];
    float abssum = (q0 + q1) + (q2 + q3);

    // halves hold hidden j and j+8 of the same column: one xor-16 combine
    abssum += __shfl_xor(abssum, 16, 32);

    const float base = fmaf(T1, a, fmaf(T2, p, T0));   // 0.25*(S0+S1*a+S2*p)+b2
    const float tot  = fmaf(0.5f, abssum, base);

    if (lane < 16 && r < nrows) out[r] = tot;
  }
}

__global__ __launch_bounds__(256) void simplenn_wmma_kernel(
    const float* __restrict__ x,     // (B, 10)
    const float* __restrict__ gate,  // (4,)
    const float* __restrict__ w1,    // (2, 64)
    const float* __restrict__ b1,    // (64,)
    const float* __restrict__ w2,    // (64, 1)
    const float* __restrict__ b2,    // (1,)
    float* __restrict__ out,         // (B*5,)
    int nrows, int ntiles)
{
  const int lane = threadIdx.x & 31;
  const int n    = lane & 15;   // column within 16-wide tile (batch row)
  const int half = lane >> 4;   // 0: K=0/1 & accum rows 0..7 ; 1: K=2/3 & rows 8..15

  // ---- uniform gate argmax (first occurrence of max, like jnp.argmax) ----
  float g0 = gate[0], g1 = gate[1], g2 = gate[2], g3 = gate[3];
  int idx = 0; float gm = g0;
  if (g1 > gm) { gm = g1; idx = 1; }
  if (g2 > gm) { gm = g2; idx = 2; }
  if (g3 > gm) { idx = 3; }

  // ---- per-wave constant operands -------------------------------------
  // Fold SIGNED w2 into W1/b1:  z''_j = w2_j * (a*W1[0,j] + p*W1[1,j] + b1_j).
  // A operand (16x4 f32): lanes 0-15: M=lane, VGPR0=K0, VGPR1=K1; lanes 16-31: K2/K3 pad.
  v2f   aop[4];       // 4 tiles cover hidden 0..63
  v8f   cinit[4];     // C operand = w2_j*b1_j in accumulator layout (v + 8*half)
  float sreg[4][8];   // sign(w2_j) in accumulator element layout
  float su = 0.0f, sv = 0.0f, sc = 0.0f;   // per-lane partials of sum_j w2.*W1, w2.*b1
#pragma unroll
  for (int t = 0; t < 4; ++t) {
    const int j   = 16 * t + n;
    const float s = w2[j];
    const float u = w1[j]       * s;    // W1[0][j] * w2_j
    const float v = w1[HID + j] * s;    // W1[1][j] * w2_j
    su += u; sv += v; sc += b1[j] * s;
    aop[t][0] = half ? 0.0f : u;        // K=0 (lanes 0-15) / K=2 pad (lanes 16-31)
    aop[t][1] = half ? 0.0f : v;        // K=1 / K=3 pad
#pragma unroll
    for (int vv = 0; vv < 8; ++vv) {
      const int jj = 16 * t + vv + 8 * half;   // hidden unit of this accum element
      const float w2x = w2[jj];
      sreg[t][vv]  = (w2x < 0.0f) ? -1.0f : 1.0f;
      cinit[t][vv] = b1[jj] * w2x;
    }
  }
  // Reduce linear-term sums over all 32 lanes (halves duplicate -> 2x true sum),
  // so scale by 0.25 to get 0.5 * sum_j.
#pragma unroll
  for (int m = 16; m >= 1; m >>= 1) {
    su += __shfl_xor(su, m, 32);
    sv += __shfl_xor(sv, m, 32);
    sc += __shfl_xor(sc, m, 32);
  }
  const float T0 = fmaf(0.25f, sc, b2[0]);
  const float T1 = 0.25f * su;
  const float T2 = 0.25f * sv;

  const int wave  = (int)((blockIdx.x * blockDim.x + threadIdx.x) >> 5);
  const int tile0 = wave * TILES_PER_WAVE;

  switch (idx) {   // uniform: one scalar dispatch per wave, loop body branch-free
    case 0:  run_tiles<0>(x, out, tile0, ntiles, nrows, lane, n, half, aop, cinit, sreg, T0, T1, T2); break;
    case 1:  run_tiles<1>(x, out, tile0, ntiles, nrows, lane, n, half, aop, cinit, sreg, T0, T1, T2); break;
    case 2:  run_tiles<2>(x, out, tile0, ntiles, nrows, lane, n, half, aop, cinit, sreg, T0, T1, T2); break;
    default: run_tiles<3>(x, out, tile0, ntiles, nrows, lane, n, half, aop, cinit, sreg, T0, T1, T2); break;
  }
}

extern "C" void kernel_launch(void* const* d_in, const int* in_sizes, int n_in,
                              void* d_out, int out_size, void* d_ws, size_t ws_size,
                              hipStream_t stream) {
  const float* x    = (const float*)d_in[0];
  const float* gate = (const float*)d_in[1];
  const float* w1   = (const float*)d_in[2];
  const float* b1   = (const float*)d_in[3];
  const float* w2   = (const float*)d_in[4];
  const float* b2   = (const float*)d_in[5];
  float* out = (float*)d_out;

  const int nrows  = out_size;                       // BATCH * 5
  const int ntiles = (nrows + 15) / 16;
  const int waves  = (ntiles + TILES_PER_WAVE - 1) / TILES_PER_WAVE;
  const long threads = (long)waves * 32;
  const int  block = 256;
  const int  grid  = (int)((threads + block - 1) / block);

  simplenn_wmma_kernel<<<grid, block, 0, stream>>>(x, gate, w1, b1, w2, b2, out,
                                                   nrows, ntiles);
}